// ModelDeep_71597104824827
// MI455X (gfx1250) — compile-verified
//
#include <hip/hip_runtime.h>
#include <hip/hip_bf16.h>
#include <math.h>

typedef __attribute__((ext_vector_type(2))) float v2f;
typedef __attribute__((ext_vector_type(8))) float v8f;

#define NNODES 50000
#define NEDGES 800000
#define DIN    128
#define DHID   128
#define DEMB   64
#define LEAKY  0.01f
#define BN_EPS 1e-5f

// ---------------------------------------------------------------------------
// small utility kernels
// ---------------------------------------------------------------------------
__global__ void init_const_kernel(float* __restrict__ p, float v, int n) {
    int i = blockIdx.x * blockDim.x + threadIdx.x;
    if (i < n) p[i] = v;
}

__global__ void deg_count_kernel(const int* __restrict__ dst,
                                 float* __restrict__ deg, int ne) {
    int e = blockIdx.x * blockDim.x + threadIdx.x;
    if (e < ne) atomicAdd(&deg[dst[e]], 1.0f);
}

__global__ void dinv_kernel(float* __restrict__ deg, int n) {
    int i = blockIdx.x * blockDim.x + threadIdx.x;
    if (i < n) {
        float d = deg[i];
        deg[i] = (d > 0.0f) ? (1.0f / sqrtf(d)) : 0.0f;
    }
}

__global__ void bias_relu_kernel(float* __restrict__ h,
                                 const float* __restrict__ b, int dim, int n) {
    int i = blockIdx.x * blockDim.x + threadIdx.x;
    if (i < n) {
        float v = h[i] + b[i % dim];
        h[i] = v > 0.0f ? v : 0.0f;
    }
}

// ---------------------------------------------------------------------------
// WMMA f32 GEMM:  C[M,N] = A[M,K] @ B[K,N]
//   - one wave computes one 16x16 output tile with V_WMMA_F32_16X16X4_F32
//   - M, N multiples of 16, K multiple of 4
//   - early-exit is wave-uniform, so EXEC is all 1s at every WMMA
// ---------------------------------------------------------------------------
__global__ void gemm_wmma_f32(const float* __restrict__ A,
                              const float* __restrict__ B,
                              float* __restrict__ C,
                              int M, int N, int K) {
    const int lane = threadIdx.x & 31;
    const int wave = threadIdx.x >> 5;
    const int wavesPerBlock = blockDim.x >> 5;
    const int ntn = N >> 4;                       // N tiles
    const int tile = blockIdx.x * wavesPerBlock + wave;
    const int ntiles = (M >> 4) * ntn;
    if (tile >= ntiles) return;                   // wave-uniform exit

    const int tm = tile / ntn;
    const int tn = tile % ntn;
    const int m0 = tm << 4;
    const int n0 = tn << 4;
    const int half = lane >> 4;                   // 0: K{0,1}, 1: K{2,3}
    const int l    = lane & 15;

    // A: row (m0+l), starting K = half*2     (ISA 7.12.2 32-bit A 16x4 layout)
    const float* arow = A + (size_t)(m0 + l) * K + (half << 1);
    // B: column (n0+l), rows half*2 + {0,1}
    const float* bcol = B + (size_t)(half << 1) * N + (n0 + l);

    v8f acc = {};
    for (int k = 0; k < K; k += 4) {
        v2f a, b;
        a.x = arow[k];
        a.y = arow[k + 1];
        b.x = bcol[(size_t)k * N];
        b.y = bcol[(size_t)(k + 1) * N];
        acc = __builtin_amdgcn_wmma_f32_16x16x4_f32(
            /*neg_a=*/false, a, /*neg_b=*/false, b,
            /*c_mod=*/(short)0, acc, /*reuse_a=*/false, /*reuse_b=*/false);
    }

    // D layout: lane<16 -> rows m0..m0+7, lane>=16 -> rows m0+8..m0+15
    const int rowBase = m0 + (half << 3);
    const int col = n0 + l;
#pragma unroll
    for (int r = 0; r < 8; ++r)
        C[(size_t)(rowBase + r) * N + col] = acc[r];
}

// ---------------------------------------------------------------------------
// Edge-parallel gather-scale-scatter:
//   agg[dst] += h[src] * dinv[src]*dinv[dst], edges [0,ne) from arrays,
//   edges [ne, ne+nn) are self-loops. dim/4 threads per edge, float4 each.
//   Random accesses stay inside L2 (node tables are 25.6/12.8 MB << 192 MB).
// ---------------------------------------------------------------------------
__global__ void edge_aggregate_kernel(const float* __restrict__ h,
                                      const int* __restrict__ src,
                                      const int* __restrict__ dst,
                                      const float* __restrict__ dinv,
                                      float* __restrict__ agg,
                                      int ne, int nn, int dim) {
    const int tpe = dim >> 2;                     // threads per edge
    const long long gid = (long long)blockIdx.x * blockDim.x + threadIdx.x;
    const long long e = gid / tpe;
    const int c = (int)(gid % tpe) << 2;
    const long long ntot = (long long)ne + nn;
    if (e >= ntot) return;

    int s, d;
    if (e < ne) { s = src[e]; d = dst[e]; }
    else        { s = (int)(e - ne); d = s; }

    const float nrm = dinv[s] * dinv[d];
    const float4 v = *(const float4*)(h + (size_t)s * dim + c);
    float* o = agg + (size_t)d * dim + c;
    atomicAdd(o + 0, v.x * nrm);
    atomicAdd(o + 1, v.y * nrm);
    atomicAdd(o + 2, v.z * nrm);
    atomicAdd(o + 3, v.w * nrm);
}

// ---------------------------------------------------------------------------
// Fused head: z = BN(leaky(agg2 + b2)); y = leaky(z@fc1^T + fc1_b);
//             out = y@fc2^T + fc2_b   (one thread per node)
// ---------------------------------------------------------------------------
__global__ void head_kernel(const float* __restrict__ agg2,
                            const float* __restrict__ b2,
                            const float* __restrict__ gamma,
                            const float* __restrict__ beta,
                            const float* __restrict__ fc1W,   // [10,64]
                            const float* __restrict__ fc1b,   // [10]
                            const float* __restrict__ fc2W,   // [5,10]
                            const float* __restrict__ fc2b,   // [5]
                            float* __restrict__ out, int n) {
    __shared__ float sW1[10 * DEMB];
    __shared__ float sW2[5 * 10];
    __shared__ float sb1[10], sb2f[5];
    __shared__ float sB2[DEMB], sG[DEMB], sBe[DEMB];

    for (int i = threadIdx.x; i < 10 * DEMB; i += blockDim.x) sW1[i] = fc1W[i];
    for (int i = threadIdx.x; i < 50;        i += blockDim.x) sW2[i] = fc2W[i];
    for (int i = threadIdx.x; i < 10;        i += blockDim.x) sb1[i] = fc1b[i];
    for (int i = threadIdx.x; i < 5;         i += blockDim.x) sb2f[i] = fc2b[i];
    for (int i = threadIdx.x; i < DEMB;      i += blockDim.x) {
        sB2[i] = b2[i]; sG[i] = gamma[i]; sBe[i] = beta[i];
    }
    __syncthreads();

    const int node = blockIdx.x * blockDim.x + threadIdx.x;
    if (node >= n) return;

    const float bnScale = rsqrtf(1.0f + BN_EPS);
    float z[DEMB];
    const float* row = agg2 + (size_t)node * DEMB;
#pragma unroll 4
    for (int k = 0; k < DEMB; ++k) {
        float v = row[k] + sB2[k];
        v = (v > 0.0f) ? v : LEAKY * v;           // leaky_relu
        z[k] = v * bnScale * sG[k] + sBe[k];      // BatchNorm1d (eval)
    }

    float y[10];
#pragma unroll
    for (int j = 0; j < 10; ++j) {
        float acc = sb1[j];
        const float* w = sW1 + j * DEMB;
        for (int k = 0; k < DEMB; ++k) acc += z[k] * w[k];
        y[j] = (acc > 0.0f) ? acc : LEAKY * acc;
    }

#pragma unroll
    for (int m = 0; m < 5; ++m) {
        float acc = sb2f[m];
        const float* w = sW2 + m * 10;
#pragma unroll
        for (int j = 0; j < 10; ++j) acc += y[j] * w[j];
        out[(size_t)node * 5 + m] = acc;
    }
}

// ---------------------------------------------------------------------------
extern "C" void kernel_launch(void* const* d_in, const int* in_sizes, int n_in,
                              void* d_out, int out_size, void* d_ws, size_t ws_size,
                              hipStream_t stream) {
    const float* x     = (const float*)d_in[0];
    const int*   eidx  = (const int*)d_in[1];       // [2, NEDGES]
    const float* W1    = (const float*)d_in[2];
    const float* b1    = (const float*)d_in[3];
    const float* W2    = (const float*)d_in[4];
    const float* b2    = (const float*)d_in[5];
    const float* gamma = (const float*)d_in[6];
    const float* beta  = (const float*)d_in[7];
    const float* fc1W  = (const float*)d_in[8];
    const float* fc1b  = (const float*)d_in[9];
    const float* fc2W  = (const float*)d_in[10];
    const float* fc2b  = (const float*)d_in[11];
    float* out = (float*)d_out;

    const int nn = in_sizes[0] / DIN;               // 50000
    const int ne = in_sizes[1] / 2;                 // 800000
    const int* src = eidx;
    const int* dst = eidx + ne;

    // workspace layout (floats, 64-aligned offsets)
    float* ws = (float*)d_ws;
    float* deg  = ws;                               // [nn]      (becomes dinv)
    float* h1   = ws + 50048;                       // [nn,128]
    float* agg1 = h1 + (size_t)nn * DHID;           // [nn,128]
    float* h2   = agg1 + (size_t)nn * DHID;         // [nn,64]
    float* agg2 = h2 + (size_t)nn * DEMB;           // [nn,64]

    const int B = 256;
    auto cdiv = [](long long a, long long b) { return (int)((a + b - 1) / b); };

    // 1) degree (self-loop contributes 1) -> dinv in place
    init_const_kernel<<<cdiv(nn, B), B, 0, stream>>>(deg, 1.0f, nn);
    init_const_kernel<<<cdiv((long long)nn * DHID, B), B, 0, stream>>>(agg1, 0.0f, nn * DHID);
    init_const_kernel<<<cdiv((long long)nn * DEMB, B), B, 0, stream>>>(agg2, 0.0f, nn * DEMB);
    deg_count_kernel<<<cdiv(ne, B), B, 0, stream>>>(dst, deg, ne);
    dinv_kernel<<<cdiv(nn, B), B, 0, stream>>>(deg, nn);

    // 2) h1 = x @ W1   (WMMA f32)
    {
        const int tiles = (nn >> 4) * (DHID >> 4);  // 3125 * 8
        gemm_wmma_f32<<<cdiv(tiles, 8), B, 0, stream>>>(x, W1, h1, nn, DHID, DIN);
    }
    // 3) agg1 = scatter(h1 * norm); then bias + relu
    {
        const long long thr = (long long)(ne + nn) * (DHID >> 2);
        edge_aggregate_kernel<<<cdiv(thr, B), B, 0, stream>>>(h1, src, dst, deg, agg1, ne, nn, DHID);
        bias_relu_kernel<<<cdiv((long long)nn * DHID, B), B, 0, stream>>>(agg1, b1, DHID, nn * DHID);
    }
    // 4) h2 = agg1 @ W2   (WMMA f32)
    {
        const int tiles = (nn >> 4) * (DEMB >> 4);  // 3125 * 4
        gemm_wmma_f32<<<cdiv(tiles, 8), B, 0, stream>>>(agg1, W2, h2, nn, DEMB, DHID);
    }
    // 5) agg2 = scatter(h2 * norm)
    {
        const long long thr = (long long)(ne + nn) * (DEMB >> 2);
        edge_aggregate_kernel<<<cdiv(thr, B), B, 0, stream>>>(h2, src, dst, deg, agg2, ne, nn, DEMB);
    }
    // 6) fused bias/leaky/BN/fc1/fc2 head
    head_kernel<<<cdiv(nn, B), B, 0, stream>>>(agg2, b2, gamma, beta,
                                               fc1W, fc1b, fc2W, fc2b, out, nn);
}